// YOLOv2Loss_86818468921632
// MI455X (gfx1250) — compile-verified
//
#include <hip/hip_runtime.h>
#include <hip/hip_bf16.h>
#include <math.h>

// Problem constants (from reference)
#define B_N 128
#define FH 26
#define FW 26
#define KCELLS (FH * FW)   // 676
#define A_N 5
#define KA (KCELLS * A_N)  // 3380
#define T_N 50
#define NC 20
#define CH 25              // 5 + NUM_CLASSES
#define IGNORE_THRESH 0.75f
#define NOOBJ_SCALE 1.0f
#define OBJ_SCALE 5.0f

__device__ __constant__ float c_anch[A_N][2] = {
    {1.3221f, 1.73145f}, {3.19275f, 4.00944f}, {5.05587f, 8.09892f},
    {9.47112f, 4.84053f}, {11.2364f, 10.0071f}};

// ---------------------------------------------------------------------------
// Kernel 1: one workgroup per image. Computes per-image partial loss.
// ---------------------------------------------------------------------------
__global__ __launch_bounds__(256) void yolo_loss_main(
    const float* __restrict__ outputs, const float* __restrict__ targets,
    float* __restrict__ partial) {
  const int b = blockIdx.x;
  const int tid = threadIdx.x;
  const float* outB = outputs + (size_t)b * CH * A_N * KCELLS;
  const float* tgtB = targets + (size_t)b * T_N * 5;

  __shared__ float s_gx1[T_N], s_gy1[T_N], s_gx2[T_N], s_gy2[T_N];
  __shared__ float s_gcx[T_N], s_gcy[T_N], s_gw[T_N], s_gh[T_N], s_garea[T_N];
  __shared__ int s_gcell[T_N], s_gaidx[T_N], s_gvalid[T_N], s_gcls[T_N];
  __shared__ int s_winner[KA];
  __shared__ float s_maxiou[KA];
  __shared__ float s_conf[KA];
  __shared__ int s_posany;
  __shared__ float s_red[256];

  if (tid == 0) s_posany = 0;
  for (int j = tid; j < KA; j += 256) s_winner[j] = -1;

  // ---- GT preprocessing (50 boxes) ----
  if (tid < T_N) {
    const int t = tid;
    float r0 = tgtB[t * 5 + 0], r1 = tgtB[t * 5 + 1];
    float r2 = tgtB[t * 5 + 2], r3 = tgtB[t * 5 + 3], r4 = tgtB[t * 5 + 4];
    int valid = (r0 + r1 + r2 + r3) > 0.0f;
    float x1 = r0 * (float)FW, y1 = r1 * (float)FH;
    float x2 = r2 * (float)FW, y2 = r3 * (float)FH;
    float cx = (x1 + x2) * 0.5f, cy = (y1 + y2) * 0.5f;
    float w = x2 - x1, h = y2 - y1;
    int cxi = (int)floorf(cx); cxi = min(max(cxi, 0), FW - 1);
    int cyi = (int)floorf(cy); cyi = min(max(cyi, 0), FH - 1);
    int cell = cyi * FW + cxi;
    float garea = w * h;
    // argmax_a IoU(anchor box centered at cell+0.5, gt)  (first-max wins)
    float best = -1.0f; int bestA = 0;
    float acx = (float)cxi + 0.5f, acy = (float)cyi + 0.5f;
    for (int a = 0; a < A_N; ++a) {
      float aw = c_anch[a][0], ah = c_anch[a][1];
      float ax1 = acx - aw * 0.5f, ay1 = acy - ah * 0.5f;
      float ax2 = acx + aw * 0.5f, ay2 = acy + ah * 0.5f;
      float xi1 = fmaxf(ax1, x1), yi1 = fmaxf(ay1, y1);
      float xi2 = fminf(ax2, x2), yi2 = fminf(ay2, y2);
      float inter = fmaxf(xi2 - xi1, 0.0f) * fmaxf(yi2 - yi1, 0.0f);
      float iou = inter / (aw * ah + garea - inter);
      if (iou > best) { best = iou; bestA = a; }
    }
    s_gx1[t] = x1; s_gy1[t] = y1; s_gx2[t] = x2; s_gy2[t] = y2;
    s_gcx[t] = cx; s_gcy[t] = cy; s_gw[t] = w; s_gh[t] = h;
    s_garea[t] = garea;
    s_gcell[t] = cell; s_gaidx[t] = bestA; s_gvalid[t] = valid;
    s_gcls[t] = (int)r4;
  }
  __syncthreads();

  // ---- last-valid-writer-wins scatter == max over t (commutative, deterministic)
  if (tid < T_N && s_gvalid[tid]) {
    atomicMax(&s_winner[s_gcell[tid] * A_N + s_gaidx[tid]], tid);
  }

  // ---- pass 1: per-anchor max IoU vs GT, conf; pos_any flag ----
  for (int j = tid; j < KA; j += 256) {
    int cell = j / A_N, a = j - cell * A_N;
    int fx = cell % FW, fy = cell / FW;
    const float* o = outB + (size_t)(a * CH) * KCELLS + cell;
    float tx = 1.0f / (1.0f + expf(-o[0 * KCELLS]));
    float ty = 1.0f / (1.0f + expf(-o[1 * KCELLS]));
    float tw = expf(o[2 * KCELLS]);
    float th = expf(o[3 * KCELLS]);
    float cf = 1.0f / (1.0f + expf(-o[4 * KCELLS]));
    float aw = c_anch[a][0], ah = c_anch[a][1];
    float px = (float)fx + tx, py = (float)fy + ty;
    float pw = aw * tw, ph = ah * th;
    float px1 = px - pw * 0.5f, py1 = py - ph * 0.5f;
    float px2 = px + pw * 0.5f, py2 = py + ph * 0.5f;
    float parea = pw * ph;
    float maxv = -1.0f;  // matches where(valid, iou, -1).max()
    for (int t = 0; t < T_N; ++t) {
      if (!s_gvalid[t]) continue;
      float xi1 = fmaxf(px1, s_gx1[t]), yi1 = fmaxf(py1, s_gy1[t]);
      float xi2 = fminf(px2, s_gx2[t]), yi2 = fminf(py2, s_gy2[t]);
      float inter = fmaxf(xi2 - xi1, 0.0f) * fmaxf(yi2 - yi1, 0.0f);
      float iou = inter / (parea + s_garea[t] - inter);
      maxv = fmaxf(maxv, iou);
    }
    s_maxiou[j] = maxv;
    s_conf[j] = cf;
    if (maxv > IGNORE_THRESH) atomicOr(&s_posany, 1);
  }
  __syncthreads();

  // ---- pass 2: accumulate loss ----
  const int posany = s_posany;
  float loss = 0.0f;
  for (int j = tid; j < KA; j += 256) {
    int t = s_winner[j];
    float cf = s_conf[j];
    if (t < 0) {
      // noobj term: (conf*im)^2 / 2, im = 0 or NOOBJ_SCALE
      float m0 = (posany && s_maxiou[j] >= IGNORE_THRESH) ? 0.0f : NOOBJ_SCALE;
      float e = cf * m0;
      loss += 0.5f * e * e;
    } else {
      int cell = j / A_N, a = j - cell * A_N;
      int fx = cell % FW, fy = cell / FW;
      // objectness: im=OBJ_SCALE, it=max_iou at this slot
      float d = cf - s_maxiou[j];
      loss += 0.5f * (OBJ_SCALE * OBJ_SCALE) * d * d;
      // box term (COORD_SCALE = 1)
      const float* o = outB + (size_t)(a * CH) * KCELLS + cell;
      float tx = 1.0f / (1.0f + expf(-o[0 * KCELLS]));
      float ty = 1.0f / (1.0f + expf(-o[1 * KCELLS]));
      float tw = expf(o[2 * KCELLS]);
      float th = expf(o[3 * KCELLS]);
      float aw = c_anch[a][0], ah = c_anch[a][1];
      float bx = tx - (s_gcx[t] - (float)fx);
      float by = ty - (s_gcy[t] - (float)fy);
      float bw = tw - s_gw[t] / aw;
      float bh = th - s_gh[t] / ah;
      loss += 0.5f * (bx * bx + by * by + bw * bw + bh * bh);
      // class cross-entropy (CLASS_SCALE = 1)
      const float* cl = o + 5 * KCELLS;
      float l[NC];
      float m = -1e30f;
      for (int c = 0; c < NC; ++c) { l[c] = cl[c * KCELLS]; m = fmaxf(m, l[c]); }
      float ssum = 0.0f;
      for (int c = 0; c < NC; ++c) ssum += expf(l[c] - m);
      float lse = m + logf(ssum);
      loss += lse - l[s_gcls[t]];
    }
  }

  // deterministic block tree reduction
  s_red[tid] = loss;
  __syncthreads();
  for (int s = 128; s > 0; s >>= 1) {
    if (tid < s) s_red[tid] += s_red[tid + s];
    __syncthreads();
  }
  if (tid == 0) partial[b] = s_red[0];
}

// ---------------------------------------------------------------------------
// Kernel 2: exact f32 reduction of 128 per-image partials via
// V_WMMA_F32_16X16X4_F32 with an all-ones B matrix (D[m][n] = sum_k A[m][k]).
// One wave32, EXEC all-ones at the WMMA as required.
// ---------------------------------------------------------------------------
typedef __attribute__((ext_vector_type(2))) float v2f;
typedef __attribute__((ext_vector_type(8))) float v8f;

__global__ __launch_bounds__(32) void yolo_loss_reduce(
    const float* __restrict__ partial, float* __restrict__ out) {
  __shared__ float s_vals[B_N];
  __shared__ float s_out[2];
  const int lane = threadIdx.x;
  for (int i = lane; i < B_N; i += 32) s_vals[i] = partial[i];  // no divergence
  __syncthreads();

  // A-matrix 16x4 f32 layout: lanes 0-15 hold (M=lane, K=0..1),
  // lanes 16-31 hold (M=lane-16, K=2..3). Element index = m*4 + k.
  int base = (lane & 15) * 4 + (lane >> 4) * 2;
  v2f a0; a0.x = s_vals[base];      a0.y = s_vals[base + 1];
  v2f a1; a1.x = s_vals[64 + base]; a1.y = s_vals[64 + base + 1];
  v2f bv; bv.x = 1.0f; bv.y = 1.0f;  // B = ones(4x16)
  v8f c = {};
  c = __builtin_amdgcn_wmma_f32_16x16x4_f32(false, a0, false, bv, (short)0, c,
                                            false, false);
  c = __builtin_amdgcn_wmma_f32_16x16x4_f32(false, a1, false, bv, (short)0, c,
                                            false, false);
  // lane 0 holds D[0..7][0] across its 8 VGPRs; lane 16 holds D[8..15][0]
  float s = c[0] + c[1] + c[2] + c[3] + c[4] + c[5] + c[6] + c[7];
  if (lane == 0) s_out[0] = s;
  if (lane == 16) s_out[1] = s;
  __syncthreads();
  if (lane == 0) out[0] = (s_out[0] + s_out[1]) * (1.0f / (float)B_N);
}

extern "C" void kernel_launch(void* const* d_in, const int* in_sizes, int n_in,
                              void* d_out, int out_size, void* d_ws,
                              size_t ws_size, hipStream_t stream) {
  const float* outputs = (const float*)d_in[0];  // (B, 125, 26, 26) f32
  const float* targets = (const float*)d_in[1];  // (B, 50, 5) f32
  float* partial = (float*)d_ws;                 // 128 floats, fully written each call
  yolo_loss_main<<<B_N, 256, 0, stream>>>(outputs, targets, partial);
  yolo_loss_reduce<<<1, 32, 0, stream>>>(partial, (float*)d_out);
}